// FourierLayer_70798240907589
// MI455X (gfx1250) — compile-verified
//
#include <hip/hip_runtime.h>
#include <math.h>

// CDNA5 / gfx1250, wave32.
typedef __attribute__((ext_vector_type(2))) float v2f;
typedef __attribute__((ext_vector_type(8))) float v8f;
typedef int v4i __attribute__((ext_vector_type(4)));
typedef __attribute__((address_space(1))) v4i as1_v4i;   // global 16B vector
typedef __attribute__((address_space(3))) v4i as3_v4i;   // LDS 16B vector

#define NT 4096          // time length
#define ND 128           // feature dim
#define NB 16            // batch
#define NF 2048          // freq rows computed (use 1..2047)
#define TOPK 7           // floor(log(2047))
#define TWO_PI 6.28318530717958647692f

#define KC 64            // k-chunk staged through LDS
#define NCHUNK (NT / KC) // 64
#define WSTRIDE 68       // padded LDS row stride (floats): conflict-free A-frag reads

// ---- workspace layout (float offsets). Total ~96.2 MiB ----
#define WS_W_RE 0
#define WS_W_IM (WS_W_RE + (size_t)NF * NT)              // 8,388,608 floats each
#define WS_X_RE (WS_W_IM + (size_t)NF * NT)
#define WS_X_IM (WS_X_RE + (size_t)NB * NF * ND)         // 4,194,304 floats each
#define WS_AMP  (WS_X_IM + (size_t)NB * NF * ND)
#define WS_PH   (WS_AMP + (size_t)NB * ND * TOPK)
#define WS_IDX  (WS_PH  + (size_t)NB * ND * TOPK)

// ---------------------------------------------------------------------------
// Kernel 1: twiddles W[f][t], row-major stride NT.
//   W_re = cos(2*pi*f*t/NT), W_im = -sin(2*pi*f*t/NT)
// Exact integer phase reduction: (f*t) mod 4096 (f*t < 2^24, exact in int).
// ---------------------------------------------------------------------------
__global__ void twiddle_kernel(float* __restrict__ wre, float* __restrict__ wim) {
    int gid = blockIdx.x * blockDim.x + threadIdx.x;
    if (gid >= NF * NT) return;
    int f = gid >> 12;
    int t = gid & (NT - 1);
    int p = (f * t) & (NT - 1);
    float theta = (float)p * (TWO_PI / (float)NT);
    float s, c;
    __sincosf(theta, &s, &c);
    wre[gid] = c;
    wim[gid] = -s;
}

// ---------------------------------------------------------------------------
// Async-stage one 64x64 twiddle chunk (re+im) into LDS.
// 256 threads x 4 iters x 2 components = 8 async b128 instructions per wave.
// Each lane copies 16B; tid+256*i -> row = ti>>4, 16B segment = (ti&15).
// ---------------------------------------------------------------------------
__device__ __forceinline__
void stage_w_chunk(const float* __restrict__ wre, const float* __restrict__ wim,
                   int fbase, int k0, float* sre, float* sim, int tid) {
    #pragma unroll
    for (int i = 0; i < 4; ++i) {
        const int ti  = tid + (i << 8);
        const int row = ti >> 4;
        const int seg = (ti & 15) << 2;          // float offset within row
        const size_t go = (size_t)(fbase + row) * NT + k0 + seg;
        const int    lo = row * WSTRIDE + seg;
        float* gre = const_cast<float*>(wre + go);
        float* gim = const_cast<float*>(wim + go);
        __builtin_amdgcn_global_load_async_to_lds_b128(
            (as1_v4i*)gre, (as3_v4i*)(sre + lo), 0, 0);
        __builtin_amdgcn_global_load_async_to_lds_b128(
            (as1_v4i*)gim, (as3_v4i*)(sim + lo), 0, 0);
    }
}

// ---------------------------------------------------------------------------
// Kernel 2: X[b,f,d] = sum_t W[f,t] * x[b,t,d]  via V_WMMA_F32_16X16X4_F32.
// Grid: (NF/64, NB), 256 threads = 8 waves; wave w -> d strip [16w, 16w+16).
// A (twiddles) is identical across waves -> staged once per chunk through LDS
// with async global->LDS b128 copies (double buffered, ASYNCcnt pipelined).
// Fragment addressing (ISA 7.12.2, 32-bit A 16x4):
//   lanes 0-15 : r = lane,    K pair 0/1 (koff=0)
//   lanes 16-31: r = lane-16, K pair 2/3 (koff=2)
// ---------------------------------------------------------------------------
__global__ __launch_bounds__(256)
void dft_wmma_kernel(const float* __restrict__ x,
                     const float* __restrict__ wre,
                     const float* __restrict__ wim,
                     float* __restrict__ xre,
                     float* __restrict__ xim) {
    __shared__ float swre[2][64 * WSTRIDE];
    __shared__ float swim[2][64 * WSTRIDE];

    const int tid   = threadIdx.x;
    const int lane  = tid & 31;
    const int wave  = tid >> 5;
    const int r     = lane & 15;
    const int kh    = lane >> 4;
    const int koff  = kh << 1;
    const int b     = blockIdx.y;
    const int fbase = blockIdx.x << 6;
    const int d0    = wave << 4;

    const float* xb = x + (size_t)b * NT * ND;

    v8f cre0 = {}, cre1 = {}, cre2 = {}, cre3 = {};
    v8f cim0 = {}, cim1 = {}, cim2 = {}, cim3 = {};

    // prologue: stage chunk 0 into buffer 0
    stage_w_chunk(wre, wim, fbase, 0, swre[0], swim[0], tid);

    for (int c = 0; c < NCHUNK; ++c) {
        const int buf = c & 1;
        const int k0  = c * KC;

        if (c + 1 < NCHUNK) {
            stage_w_chunk(wre, wim, fbase, k0 + KC, swre[buf ^ 1], swim[buf ^ 1], tid);
            __builtin_amdgcn_s_wait_asynccnt(8);   // chunk c done; c+1 in flight
        } else {
            __builtin_amdgcn_s_wait_asynccnt(0);
        }
        __syncthreads();                           // chunk c visible to all waves

        const float* sre = swre[buf];
        const float* sim = swim[buf];

        // prefetch next chunk of x (B operand stream)
        if (c + 1 < NCHUNK)
            __builtin_prefetch(xb + (size_t)(k0 + KC + koff) * ND + d0 + r, 0, 1);

        #pragma unroll 4
        for (int kl = 0; kl < KC; kl += 4) {
            const int kk = kl + koff;

            // B fragment: x[b, k0+kk .. +1, d0+r]
            const float* xp = xb + (size_t)(k0 + kk) * ND + d0 + r;
            v2f bf;
            bf.x = xp[0];
            bf.y = xp[ND];

            // A fragments from LDS: 4 freq tiles x (re, im), ds_load_b64 each
            const int ro = r * WSTRIDE + kk;
            v2f ar0 = *(const v2f*)(sre + ro);
            v2f ai0 = *(const v2f*)(sim + ro);
            v2f ar1 = *(const v2f*)(sre + 16 * WSTRIDE + ro);
            v2f ai1 = *(const v2f*)(sim + 16 * WSTRIDE + ro);
            v2f ar2 = *(const v2f*)(sre + 32 * WSTRIDE + ro);
            v2f ai2 = *(const v2f*)(sim + 32 * WSTRIDE + ro);
            v2f ar3 = *(const v2f*)(sre + 48 * WSTRIDE + ro);
            v2f ai3 = *(const v2f*)(sim + 48 * WSTRIDE + ro);

            cre0 = __builtin_amdgcn_wmma_f32_16x16x4_f32(false, ar0, false, bf, (short)0, cre0, false, false);
            cim0 = __builtin_amdgcn_wmma_f32_16x16x4_f32(false, ai0, false, bf, (short)0, cim0, false, false);
            cre1 = __builtin_amdgcn_wmma_f32_16x16x4_f32(false, ar1, false, bf, (short)0, cre1, false, false);
            cim1 = __builtin_amdgcn_wmma_f32_16x16x4_f32(false, ai1, false, bf, (short)0, cim1, false, false);
            cre2 = __builtin_amdgcn_wmma_f32_16x16x4_f32(false, ar2, false, bf, (short)0, cre2, false, false);
            cim2 = __builtin_amdgcn_wmma_f32_16x16x4_f32(false, ai2, false, bf, (short)0, cim2, false, false);
            cre3 = __builtin_amdgcn_wmma_f32_16x16x4_f32(false, ar3, false, bf, (short)0, cre3, false, false);
            cim3 = __builtin_amdgcn_wmma_f32_16x16x4_f32(false, ai3, false, bf, (short)0, cim3, false, false);
        }

        __syncthreads();                           // done reading buf before reuse
    }

    // C/D layout: VGPR j -> M = j + 8*kh, N = r  (lanes 0-15: M=j, 16-31: M=j+8)
    const int d = d0 + r;
    #pragma unroll
    for (int j = 0; j < 8; ++j) {
        const int m = j + (kh << 3);
        const size_t o = ((size_t)b * NF + (fbase + m)) * ND + d;
        xre[o]                   = cre0[j];
        xim[o]                   = cim0[j];
        xre[o + (size_t)16 * ND] = cre1[j];
        xim[o + (size_t)16 * ND] = cim1[j];
        xre[o + (size_t)32 * ND] = cre2[j];
        xim[o + (size_t)32 * ND] = cim2[j];
        xre[o + (size_t)48 * ND] = cre3[j];
        xim[o + (size_t)48 * ND] = cim3[j];
    }
}

// ---------------------------------------------------------------------------
// Kernel 3: top-7 |X|^2 over f in [1, 2047] per (b,d) column.
// One wave32 per column; lane-local top-7, then LDS merge by lane 0.
// ---------------------------------------------------------------------------
__global__ __launch_bounds__(256)
void topk_kernel(const float* __restrict__ xre, const float* __restrict__ xim,
                 float* __restrict__ amp2, float* __restrict__ ph,
                 int* __restrict__ idxs) {
    __shared__ float sv[256 * TOPK];
    __shared__ int   si[256 * TOPK];

    const int tid     = threadIdx.x;
    const int lane    = tid & 31;
    const int waveBlk = tid >> 5;
    const int col     = blockIdx.x * 8 + waveBlk;   // 0 .. NB*ND-1
    const int b       = col >> 7;
    const int d       = col & (ND - 1);

    float bv[TOPK];
    int   bi[TOPK];
    #pragma unroll
    for (int k = 0; k < TOPK; ++k) { bv[k] = -1.0f; bi[k] = 1; }

    const float* pre = xre + (size_t)b * NF * ND + d;
    const float* pim = xim + (size_t)b * NF * ND + d;

    for (int f = 1 + lane; f < NF; f += 32) {
        float re = pre[(size_t)f * ND];
        float im = pim[(size_t)f * ND];
        float m  = re * re + im * im;
        int   mk = 0; float mv = bv[0];
        #pragma unroll
        for (int k = 1; k < TOPK; ++k) if (bv[k] < mv) { mv = bv[k]; mk = k; }
        if (m > mv) { bv[mk] = m; bi[mk] = f; }
    }

    #pragma unroll
    for (int k = 0; k < TOPK; ++k) { sv[tid * TOPK + k] = bv[k]; si[tid * TOPK + k] = bi[k]; }
    __syncthreads();

    if (lane == 0) {
        const int base = waveBlk * 32 * TOPK;
        for (int k = 0; k < TOPK; ++k) {
            float best = -2.0f; int bj = base;
            for (int j = 0; j < 32 * TOPK; ++j) {
                float v = sv[base + j];
                if (v > best) { best = v; bj = base + j; }
            }
            sv[bj] = -3.0f;                      // remove from candidate pool
            int f = si[bj];
            float re = pre[(size_t)f * ND];
            float im = pim[(size_t)f * ND];
            float a  = sqrtf(re * re + im * im);
            float p  = atan2f(im, re);
            size_t o = (size_t)col * TOPK + k;
            amp2[o] = 2.0f * a;                  // conjugate pair doubles the cosine
            ph[o]   = p;
            idxs[o] = f;
        }
    }
}

// ---------------------------------------------------------------------------
// Kernel 4: out[b,t,d] = sum_k 2*amp_k * cos(2*pi*(f_k*t mod NT)/NT + phi_k)
// ---------------------------------------------------------------------------
__global__ void synth_kernel(const float* __restrict__ amp2,
                             const float* __restrict__ ph,
                             const int* __restrict__ idxs,
                             float* __restrict__ out) {
    int gid = blockIdx.x * blockDim.x + threadIdx.x;
    if (gid >= NB * NT * ND) return;
    int b   = gid >> 19;              // NT*ND = 2^19
    int rem = gid & ((1 << 19) - 1);
    int t   = rem >> 7;
    int d   = rem & (ND - 1);
    int base = (b * ND + d) * TOPK;

    float acc = 0.0f;
    #pragma unroll
    for (int k = 0; k < TOPK; ++k) {
        float a2 = amp2[base + k];
        float p  = ph[base + k];
        int   f  = idxs[base + k];
        int   pi = (f * t) & (NT - 1);        // exact phase reduction
        float arg = (float)pi * (TWO_PI / (float)NT) + p;
        acc += a2 * cosf(arg);
    }
    out[gid] = acc;
}

// ---------------------------------------------------------------------------
extern "C" void kernel_launch(void* const* d_in, const int* in_sizes, int n_in,
                              void* d_out, int out_size, void* d_ws, size_t ws_size,
                              hipStream_t stream) {
    (void)in_sizes; (void)n_in; (void)out_size; (void)ws_size;

    const float* x = (const float*)d_in[0];
    float* ws  = (float*)d_ws;
    float* wre = ws + WS_W_RE;
    float* wim = ws + WS_W_IM;
    float* xre = ws + WS_X_RE;
    float* xim = ws + WS_X_IM;
    float* amp = ws + WS_AMP;
    float* phs = ws + WS_PH;
    int*   idx = (int*)(ws + WS_IDX);
    float* out = (float*)d_out;

    twiddle_kernel<<<(NF * NT + 255) / 256, 256, 0, stream>>>(wre, wim);

    dim3 g(NF / 64, NB);
    dft_wmma_kernel<<<g, 256, 0, stream>>>(x, wre, wim, xre, xim);

    topk_kernel<<<(NB * ND) / 8, 256, 0, stream>>>(xre, xim, amp, phs, idx);

    synth_kernel<<<(NB * NT * ND + 255) / 256, 256, 0, stream>>>(amp, phs, idx, out);
}